// ScaledDotProductAttention_88235808129607
// MI455X (gfx1250) — compile-verified
//
#include <hip/hip_runtime.h>
#include <stdint.h>

// Scaled dot-product attention forward for B=4,H=16,S=2048,D=64 (fp32).
// Outputs: [out (B*H*S*D) | attn logits (B*H*S*S)] concatenated in d_out.
// One wave32 per 16-query tile; flash-attention online softmax; matmuls via
// V_WMMA_F32_16X16X4_F32; K/V tiles double-buffered in LDS via
// GLOBAL_LOAD_ASYNC_TO_LDS_B128 (ASYNCcnt) when the toolchain exposes it.

namespace {

constexpr int   kS = 2048;
constexpr int   kD = 64;
constexpr int   kH = 16;
constexpr int   kB = 4;
constexpr int   kNT = kS / 16;               // 128 key tiles
constexpr float kInvTemp = 0.125f;           // 1/sqrt(64)
constexpr float kNegInf  = -1.0e9f;
constexpr float kLog2e   = 1.4426950408889634f;

constexpr int kKStride = 68;                 // padded LDS row stride (floats) for K/V tiles
constexpr int kPStride = 17;                 // padded LDS row stride for P tile

typedef float v2f __attribute__((ext_vector_type(2)));
typedef float v8f __attribute__((ext_vector_type(8)));
typedef int   v4i __attribute__((vector_size(4 * sizeof(int))));  // matches builtin param

#if defined(__gfx1250__) && __has_builtin(__builtin_amdgcn_global_load_async_to_lds_b128) && __has_builtin(__builtin_amdgcn_s_wait_asynccnt)
#define USE_ASYNC_STAGE 1
#else
#define USE_ASYNC_STAGE 0
#endif

#if USE_ASYNC_STAGE
__device__ __forceinline__ void async_cp16(const float* g, float* l) {
  // Builtin takes generic pointers to 16-byte int vectors:
  // (global src, LDS dst, imm offset, imm cpol).
  __builtin_amdgcn_global_load_async_to_lds_b128(
      (v4i*)g, (v4i*)l, /*imm offset*/0, /*cpol*/0);
}
#endif

// Issue (async) or perform (sync fallback) the copy of one 16x64 K tile and
// one 16x64 V tile into padded LDS buffers. 16 b128 transfers per wave.
__device__ __forceinline__ void stage_tile(const float* gK, const float* gV,
                                           float* dK, float* dV,
                                           int hi, int ln) {
  const int col = 4 * ln;
#if USE_ASYNC_STAGE
  #pragma unroll
  for (int it = 0; it < 8; ++it) {
    const int row = 2 * it + hi;
    async_cp16(gK + row * kD + col, dK + row * kKStride + col);
  }
  #pragma unroll
  for (int it = 0; it < 8; ++it) {
    const int row = 2 * it + hi;
    async_cp16(gV + row * kD + col, dV + row * kKStride + col);
  }
#else
  #pragma unroll
  for (int it = 0; it < 8; ++it) {
    const int row = 2 * it + hi;
    *(float4*)(dK + row * kKStride + col) = *(const float4*)(gK + row * kD + col);
    *(float4*)(dV + row * kKStride + col) = *(const float4*)(gV + row * kD + col);
  }
#endif
}

__device__ __forceinline__ float half_reduce_max(float x) {
  // xor masks 1..8 stay within each 16-lane half of the wave32:
  // reduces over the 16 columns (keys) a C-tile row occupies.
  #pragma unroll
  for (int m = 8; m >= 1; m >>= 1) x = fmaxf(x, __shfl_xor(x, m, 32));
  return x;
}
__device__ __forceinline__ float half_reduce_sum(float x) {
  #pragma unroll
  for (int m = 8; m >= 1; m >>= 1) x += __shfl_xor(x, m, 32);
  return x;
}

__global__ __launch_bounds__(32)
void sdpa_fwd(const float* __restrict__ q,
              const float* __restrict__ k,
              const float* __restrict__ v,
              const int*   __restrict__ mask,
              const float* __restrict__ bias,
              float* __restrict__ out,
              float* __restrict__ attn)
{
  __shared__ float sK[2][16 * kKStride];   // double-buffered K tiles
  __shared__ float sV[2][16 * kKStride];   // double-buffered V tiles
  __shared__ float sP[16 * kPStride];      // P round-trip (C -> A layout)

  const int lane = (int)(threadIdx.x & 31u);
  const int hi   = lane >> 4;          // which 16-lane half
  const int ln   = lane & 15;          // lane-in-half

  const int wid = (int)blockIdx.x;
  const int mt  = wid & (kNT - 1);     // query tile index within (b,h)
  const int bh  = wid >> 7;            // kNT == 128
  const int b   = bh >> 4;             // / kH

  const size_t qkv_base  = (size_t)bh * kS * kD;
  const size_t bias_base = (size_t)bh * kS * kS;
  const size_t mask_base = (size_t)b  * kS * kS;
  const int m0 = mt * 16;

  // ---- Q tile into WMMA-A (16x4 f32) layout registers, pre-scaled --------
  // A layout (ISA 7.12.2): lane L holds row M=L%16; VGPR j holds K=2*(L/16)+j.
  // Chunk c covers K-dim (=D) slice [4c, 4c+4).
  v2f aQ[16];
  {
    const float* qrow = q + qkv_base + (size_t)(m0 + ln) * kD + 2 * hi;
    #pragma unroll
    for (int c = 0; c < 16; ++c) {
      aQ[c].x = qrow[4 * c + 0] * kInvTemp;
      aQ[c].y = qrow[4 * c + 1] * kInvTemp;
    }
  }

  // Output accumulators: 4 column groups of 16, each a 16x16 f32 C tile.
  v8f accO[4];
  #pragma unroll
  for (int g = 0; g < 4; ++g) {
    #pragma unroll
    for (int r = 0; r < 8; ++r) accO[g][r] = 0.0f;
  }
  // Per-lane softmax state for rows m = r + 8*hi.
  float mrun[8], lrun[8];
  #pragma unroll
  for (int r = 0; r < 8; ++r) { mrun[r] = -1.0e30f; lrun[r] = 0.0f; }

  const float* bias_rows = bias + bias_base + (size_t)m0 * kS;
  const int*   mask_rows = mask + mask_base + (size_t)m0 * kS;
  float*       attn_rows = attn + bias_base + (size_t)m0 * kS;

  const float* kg0 = k + qkv_base;
  const float* vg0 = v + qkv_base;

  // Prologue: start staging tile 0 into buffer 0.
  stage_tile(kg0, vg0, sK[0], sV[0], hi, ln);

  for (int kt = 0; kt < kNT; ++kt) {
    const int    cur = kt & 1;
    const int    k0  = kt * 16;
    const float* sKc = sK[cur];
    const float* sVc = sV[cur];

    __syncthreads();  // drain prior DS reads before buffers get rewritten

    // Kick off next tile into the other buffer, then wait for current tile.
    // Async transfers complete in order, so asynccnt<=16 => tile kt resident.
    if (kt + 1 < kNT) {
      stage_tile(kg0 + (size_t)(k0 + 16) * kD, vg0 + (size_t)(k0 + 16) * kD,
                 sK[cur ^ 1], sV[cur ^ 1], hi, ln);
#if USE_ASYNC_STAGE
      __builtin_amdgcn_s_wait_asynccnt(16);
#endif
    } else {
#if USE_ASYNC_STAGE
      __builtin_amdgcn_s_wait_asynccnt(0);
#endif
    }
    asm volatile("" ::: "memory");

    // ---- prefetch mask/bias tile into registers (hides under WMMA chain) -
    int   mk8[8];
    float bb8[8];
    #pragma unroll
    for (int r = 0; r < 8; ++r) {
      const size_t off = (size_t)(r + 8 * hi) * kS + (size_t)(k0 + ln);
      mk8[r] = __builtin_nontemporal_load(mask_rows + off);
      bb8[r] = __builtin_nontemporal_load(bias_rows + off);
    }

    // ---- S = (Q/temp) * K^T : two independent 8-deep f32 WMMA chains -----
    v8f s8a, s8b;
    #pragma unroll
    for (int r = 0; r < 8; ++r) { s8a[r] = 0.0f; s8b[r] = 0.0f; }
    #pragma unroll
    for (int c = 0; c < 8; ++c) {
      // B layout (4x16): lane L holds col N=L%16; VGPR j holds K=2*(L/16)+j.
      v2f bK0, bK1;
      bK0.x = sKc[ln * kKStride + 8 * c + 2 * hi + 0];
      bK0.y = sKc[ln * kKStride + 8 * c + 2 * hi + 1];
      bK1.x = sKc[ln * kKStride + 8 * c + 4 + 2 * hi + 0];
      bK1.y = sKc[ln * kKStride + 8 * c + 4 + 2 * hi + 1];
      s8a = __builtin_amdgcn_wmma_f32_16x16x4_f32(
          false, aQ[2 * c + 0], false, bK0, (short)0, s8a, false, false);
      s8b = __builtin_amdgcn_wmma_f32_16x16x4_f32(
          false, aQ[2 * c + 1], false, bK1, (short)0, s8b, false, false);
    }
    const v8f s8 = s8a + s8b;

    // ---- mask, bias, emit logits (non-temporal stream) -------------------
    float sv[8];
    #pragma unroll
    for (int r = 0; r < 8; ++r) {
      const size_t off = (size_t)(r + 8 * hi) * kS + (size_t)(k0 + ln);
      const float val = (mk8[r] == 0 ? kNegInf : s8[r]) + bb8[r];
      __builtin_nontemporal_store(val, attn_rows + off);
      sv[r] = val;
    }

    // ---- online softmax update ------------------------------------------
    float p[8];
    #pragma unroll
    for (int r = 0; r < 8; ++r) {
      const float tmax = half_reduce_max(sv[r]);
      const float mnew = fmaxf(mrun[r], tmax);
      p[r] = __builtin_amdgcn_exp2f((sv[r] - mnew) * kLog2e);
      const float psum  = half_reduce_sum(p[r]);
      const float alpha = __builtin_amdgcn_exp2f((mrun[r] - mnew) * kLog2e);
      lrun[r] = lrun[r] * alpha + psum;
      mrun[r] = mnew;
      #pragma unroll
      for (int g = 0; g < 4; ++g) accO[g][r] *= alpha;
    }

    // ---- P through LDS: C layout -> A layout -----------------------------
    #pragma unroll
    for (int r = 0; r < 8; ++r) sP[(r + 8 * hi) * kPStride + ln] = p[r];
    __syncthreads();

    // ---- O += P * V : 4 key sub-chunks x 4 D-column groups ---------------
    #pragma unroll
    for (int c = 0; c < 4; ++c) {
      v2f aP;
      aP.x = sP[ln * kPStride + 4 * c + 2 * hi + 0];
      aP.y = sP[ln * kPStride + 4 * c + 2 * hi + 1];
      #pragma unroll
      for (int g = 0; g < 4; ++g) {
        v2f bV;
        bV.x = sVc[(4 * c + 2 * hi + 0) * kKStride + 16 * g + ln];
        bV.y = sVc[(4 * c + 2 * hi + 1) * kKStride + 16 * g + ln];
        accO[g] = __builtin_amdgcn_wmma_f32_16x16x4_f32(
            false, aP, false, bV, (short)0, accO[g], false, false);
      }
    }
  }

  // ---- finalize: O /= l, store (each C row = 64B contiguous per half) ----
  float* orow = out + qkv_base + (size_t)m0 * kD;
  #pragma unroll
  for (int r = 0; r < 8; ++r) {
    const float invl = 1.0f / lrun[r];
    #pragma unroll
    for (int g = 0; g < 4; ++g) {
      __builtin_nontemporal_store(
          accO[g][r] * invl,
          orow + (size_t)(r + 8 * hi) * kD + 16 * g + ln);
    }
  }
}

}  // namespace

extern "C" void kernel_launch(void* const* d_in, const int* in_sizes, int n_in,
                              void* d_out, int out_size, void* d_ws, size_t ws_size,
                              hipStream_t stream) {
  const float* q    = (const float*)d_in[0];
  const float* k    = (const float*)d_in[1];
  const float* v    = (const float*)d_in[2];
  const int*   mask = (const int*)d_in[3];
  const float* bias = (const float*)d_in[4];

  float* out  = (float*)d_out;
  float* attn = out + (size_t)kB * kH * kS * kD;   // second tuple element

  const int num_waves = kB * kH * (kS / 16);       // 8192 query tiles
  sdpa_fwd<<<dim3(num_waves), dim3(32), 0, stream>>>(q, k, v, mask, bias, out, attn);

  (void)in_sizes; (void)n_in; (void)out_size; (void)d_ws; (void)ws_size;
}